// ConvSP_52742198395417
// MI455X (gfx1250) — compile-verified
//
#include <hip/hip_runtime.h>

// ConvSP on MI455X (gfx1250, wave32, WMMA).
// out[b,i,o] = bias[o] + sum_{k,j} W_k[i,j] * u[b,k,j,o]
//   u[b,k,j,o] = sum_c data[b,j,c]/(invmass*density) * weight[o,c,k]
//   W_k[i,j]   = cubic-spline kernel of || pos_i + off_k - pos_j ||
// u stored pair-interleaved: u2[b][k][j>>1][o][j&1]  (so B-fragment = 1x b64 load)

#define NN      4096
#define BB      2
#define CC      32
#define OO      32
#define NCELLS  27
#define JSPLIT  16          // split j-range per (b,itile) for occupancy
#define SIGMA   2546.4790894703254f   // 8 / (pi * h^3), h = 0.1

typedef __attribute__((ext_vector_type(2))) float v2f;
typedef __attribute__((ext_vector_type(8))) float v8f;

// ---------------------------------------------------------------- u precompute
__global__ void __launch_bounds__(256) convsp_u_kernel(
    const float* __restrict__ locs, const float* __restrict__ data,
    const float* __restrict__ density, const float* __restrict__ weight,
    float* __restrict__ u) {
  const int idx = blockIdx.x * 256 + threadIdx.x;   // over N*O
  const int o = idx & (OO - 1);
  const int j = idx >> 5;
  const int k = blockIdx.y;
  const int b = blockIdx.z;
  const float invm  = locs[(b * NN + j) * 4 + 3];
  const float scale = 1.0f / (invm * density[b * NN + j]);
  const float* __restrict__ drow = data + (size_t)(b * NN + j) * CC;
  float sum = 0.0f;
#pragma unroll
  for (int c = 0; c < CC; ++c)
    sum = fmaf(drow[c], weight[(o * CC + c) * NCELLS + k], sum);
  // pair-interleaved layout: [b][k][j>>1][o][j&1]
  u[((((size_t)b * NCELLS + k) * (NN / 2) + (j >> 1)) * OO + o) * 2 + (j & 1)] =
      sum * scale;
}

// ---------------------------------------------------------------- out = bias
__global__ void __launch_bounds__(256) convsp_init_kernel(
    const float* __restrict__ bias, float* __restrict__ out) {
  const int idx = blockIdx.x * 256 + threadIdx.x;
  out[idx] = bias[idx & (OO - 1)];
}

// ---------------------------------------------------------------- SPH weight
__device__ __forceinline__ float kernel_w(float d2) {
  d2 = fmaxf(d2, 0.0f);
  const float d  = __builtin_sqrtf(d2);
  const float q  = d * 10.0f;                  // q = d / h, h = 0.1
  const float q2 = q * q;
  const float w1 = fmaf(6.0f * q, q2, fmaf(-6.0f, q2, 1.0f)); // 1 - 6q^2 + 6q^3
  const float om = 1.0f - q;
  const float w2 = 2.0f * om * om * om;
  float w = (q <= 0.5f) ? w1 : w2;
  return (q < 1.0f) ? SIGMA * w : 0.0f;
}

// ---------------------------------------------------------------- main GEMM
// One wave -> 16 rows (i) x 32 cols (o) for one (b, itile, jchunk).
// A fragment (16x4 f32): lane half h = lane>>4, row m = lane&15, K = 2h + v.
// B fragment (4x16 f32): col n = lane&15, K = 2h + v.
__global__ void __launch_bounds__(256) convsp_main_kernel(
    const float* __restrict__ locs, const float* __restrict__ u,
    float* __restrict__ out) {
  const int lane = threadIdx.x & 31;
  const int h    = lane >> 4;        // lane half (selects K pair / M high rows)
  const int m    = lane & 15;        // A row within tile; B/D column
  const int wid  = blockIdx.x * 8 + (threadIdx.x >> 5);
  const int b      = wid >> 12;            // 4096 waves per batch
  const int rem    = wid & 4095;
  const int itile  = rem >> 4;             // 256 i-tiles
  const int jchunk = rem & (JSPLIT - 1);   // JSPLIT chunks of 256 j's
  const int i0     = itile * 16;
  const int JCH    = NN / JSPLIT;          // 256 j's per chunk

  const float4* __restrict__ locs4 = (const float4*)locs;
  const float4 li = locs4[b * NN + i0 + m];
  const float* __restrict__ ub = u + (size_t)b * NCELLS * NN * OO;

  v8f acc0 = {};   // o = 0..15
  v8f acc1 = {};   // o = 16..31

  for (int jt = 0; jt < JCH / 16; ++jt) {
    const int j0 = jchunk * JCH + jt * 16;

    // Stage the 8 (i,j) pair deltas this lane owns (reused for all 27 cells).
    float dx[8], dy[8], dz[8], dd[8];
#pragma unroll
    for (int p = 0; p < 8; ++p) {
      const int jj = 4 * (p >> 1) + 2 * h + (p & 1);
      const float4 lj = locs4[b * NN + j0 + jj];
      const float ddx = li.x - lj.x;
      const float ddy = li.y - lj.y;
      const float ddz = li.z - lj.z;
      dx[p] = ddx; dy[p] = ddy; dz[p] = ddz;
      dd[p] = fmaf(ddx, ddx, fmaf(ddy, ddy, ddz * ddz));
    }
    if (jt + 1 < JCH / 16)
      __builtin_prefetch(&locs4[b * NN + j0 + 16], 0, 3);

    for (int k = 0; k < NCELLS; ++k) {
      // off_k = (g - 1) * 0.05, g = (k/9, (k/3)%3, k%3); need 2*off and |off|^2
      const float tx = 0.1f * (float)(k / 9 - 1);
      const float ty = 0.1f * (float)((k / 3) % 3 - 1);
      const float tz = 0.1f * (float)(k % 3 - 1);
      const float cst = 0.25f * (tx * tx + ty * ty + tz * tz);
      // per-k u tile base; pair-interleaved so j0*OO still holds (j0 even)
      const float* __restrict__ uk = ub + (size_t)k * NN * OO + (size_t)j0 * OO;

      // Prefetch next cell's 2KB u tile (lanes stride one 64B line each).
      if (k + 1 < NCELLS)
        __builtin_prefetch(uk + (size_t)NN * OO + lane * 16, 0, 3);

#pragma unroll
      for (int s = 0; s < 4; ++s) {
        // Two A-fragment values: W[i=m][j0 + 4s + 2h + {0,1}]
        const int p0 = 2 * s, p1 = 2 * s + 1;
        const float d20 = fmaf(tx, dx[p0], fmaf(ty, dy[p0], fmaf(tz, dz[p0], dd[p0] + cst)));
        const float d21 = fmaf(tx, dx[p1], fmaf(ty, dy[p1], fmaf(tz, dz[p1], dd[p1] + cst)));
        v2f a = { kernel_w(d20), kernel_w(d21) };

        // B fragments: interleaved rows (jA, jA+1), cols m and m+16 -> b64 loads
        const int jA = 4 * s + 2 * h;                 // even
        const float* fb = uk + jA * OO;               // start of interleaved pair
        v2f b0 = *(const v2f*)(fb + 2 * m);
        v2f b1 = *(const v2f*)(fb + 2 * (16 + m));

        acc0 = __builtin_amdgcn_wmma_f32_16x16x4_f32(
            false, a, false, b0, (short)0, acc0, false, false);
        acc1 = __builtin_amdgcn_wmma_f32_16x16x4_f32(
            false, a, false, b1, (short)0, acc1, false, false);
      }
    }
  }

  // D layout: VGPR r, lanes 0-15 -> row i0+r, lanes 16-31 -> row i0+r+8; col m.
#pragma unroll
  for (int r = 0; r < 8; ++r) {
    const int row = i0 + r + 8 * h;
    float* dst = out + ((size_t)b * NN + row) * OO + m;
    unsafeAtomicAdd(dst,      acc0[r]);
    unsafeAtomicAdd(dst + 16, acc1[r]);
  }
}

// ---------------------------------------------------------------- launch
extern "C" void kernel_launch(void* const* d_in, const int* in_sizes, int n_in,
                              void* d_out, int out_size, void* d_ws, size_t ws_size,
                              hipStream_t stream) {
  const float* locs    = (const float*)d_in[0];  // (B,N,4) xyz + invmass
  const float* data    = (const float*)d_in[1];  // (B,N,C)
  const float* density = (const float*)d_in[2];  // (B,N)
  const float* weight  = (const float*)d_in[3];  // (O,C,27)
  const float* bias    = (const float*)d_in[4];  // (O)
  float* out = (float*)d_out;                    // (B,N,O)
  float* u   = (float*)d_ws;                     // (B,27,N/2,O,2) = ~27 MB

  dim3 gu(NN * OO / 256, NCELLS, BB);
  convsp_u_kernel<<<gu, 256, 0, stream>>>(locs, data, density, weight, u);

  convsp_init_kernel<<<(BB * NN * OO) / 256, 256, 0, stream>>>(bias, out);

  // 8192 waves total = BB * (NN/16) * JSPLIT ; 8 waves per block.
  convsp_main_kernel<<<BB * (NN / 16) * JSPLIT / 8, 256, 0, stream>>>(locs, u, out);
}